// GeometricUnit_12206297055627
// MI455X (gfx1250) — compile-verified
//
#include <hip/hip_runtime.h>
#include <math.h>

typedef __attribute__((ext_vector_type(2))) float v2f;
typedef __attribute__((ext_vector_type(8))) float v8f;

#define N_PTS 4096
#define N_BATCH 8
#define KNN 4
#define TILE 16
#define NTILES (N_PTS / TILE)          // 256
#define WAVES_PER_BLOCK 4
#define ROWS_PER_WAVE 16
#define ROWS_PER_BLOCK (WAVES_PER_BLOCK * ROWS_PER_WAVE)  // 64
#define EPSV 1e-5f
#define SLOPE 0.1f

// ---------------------------------------------------------------------------
// Kernel 1: per-batch KNN (via WMMA pdist tiles) -> laplacian feature -> 3x3
// transform. Writes trans[B][N][3] to workspace.
// ---------------------------------------------------------------------------
__global__ __launch_bounds__(128) void knn_feat_kernel(
    const float* __restrict__ xyz,   // [B][N][3]
    const float* __restrict__ Wm,    // [3][3]
    const float* __restrict__ bias,  // [3]
    float* __restrict__ trans)       // [B][N][3] (workspace)
{
    // 32KB shared: phase A = x[4096], y[4096] staging;
    // phase B (after barrier) = merge buffers mD/mI (4*16*64 each).
    __shared__ float smem[2 * N_PTS];
    float* sx = smem;
    float* sy = smem + N_PTS;
    float* mD = smem;                              // 4096 floats
    int*   mI = (int*)(smem + N_PTS);              // 4096 ints

    const int b    = blockIdx.y;
    const int tid  = threadIdx.x;
    const int wave = tid >> 5;
    const int lane = tid & 31;
    const int sub  = lane & 15;
    const bool hi  = (lane >= 16);

    const float* xb = xyz + (size_t)b * N_PTS * 3;

    // ---- stage x,y into LDS (uniform trip count; EXEC stays full) ----
    for (int i = tid; i < N_PTS; i += 128) {
        sx[i] = xb[i * 3 + 0];
        sy[i] = xb[i * 3 + 1];
    }
    __syncthreads();

    const int qbase = blockIdx.x * ROWS_PER_BLOCK + wave * ROWS_PER_WAVE;

    // ---- A matrix (16x4 f32): row m = qbase+sub.
    // lanes 0-15 : VGPR0=K0=-2x, VGPR1=K1=-2y
    // lanes 16-31: VGPR0=K2=1,   VGPR1=K3=xx
    const float qx = sx[qbase + sub];
    const float qy = sy[qbase + sub];
    v2f amat;
    amat.x = hi ? 1.0f : (-2.0f * qx);
    amat.y = hi ? (qx * qx + qy * qy) : (-2.0f * qy);

    // running sorted top-4 (ascending) per 8 rows
    float bd[8][KNN];
    int   bi[8][KNN];
#pragma unroll
    for (int r = 0; r < 8; ++r)
#pragma unroll
        for (int k = 0; k < KNN; ++k) { bd[r][k] = 3.0e38f; bi[r][k] = 0; }

    const v8f cz = {0.f, 0.f, 0.f, 0.f, 0.f, 0.f, 0.f, 0.f};

    // ---- sweep all 256 column tiles ----
    for (int t = 0; t < NTILES; ++t) {
        const int c = t * TILE + sub;
        const float xc = sx[c];
        const float yc = sy[c];
        // B matrix (4x16 f32): lanes 0-15: K0=x,K1=y ; lanes 16-31: K2=xx,K3=1
        v2f bmat;
        bmat.x = hi ? (xc * xc + yc * yc) : xc;
        bmat.y = hi ? 1.0f : yc;

        v8f d = __builtin_amdgcn_wmma_f32_16x16x4_f32(
            false, amat, false, bmat, (short)0, cz, false, false);

        // lane l, elem r -> pdist(row r+8*hi, col c); insert (d[r], c)
#pragma unroll
        for (int r = 0; r < 8; ++r) {
            const float v = d[r];
            const bool c0 = v < bd[r][0];
            const bool c1 = v < bd[r][1];
            const bool c2 = v < bd[r][2];
            const bool c3 = v < bd[r][3];
            bd[r][3] = c2 ? bd[r][2] : (c3 ? v : bd[r][3]);
            bi[r][3] = c2 ? bi[r][2] : (c3 ? c : bi[r][3]);
            bd[r][2] = c1 ? bd[r][1] : (c2 ? v : bd[r][2]);
            bi[r][2] = c1 ? bi[r][1] : (c2 ? c : bi[r][2]);
            bd[r][1] = c0 ? bd[r][0] : (c1 ? v : bd[r][1]);
            bi[r][1] = c0 ? bi[r][0] : (c1 ? c : bi[r][1]);
            bd[r][0] = c0 ? v : bd[r][0];
            bi[r][0] = c0 ? c : bi[r][0];
        }
    }

    // staging buffers no longer needed; reuse LDS for merge
    __syncthreads();

#pragma unroll
    for (int r = 0; r < 8; ++r) {
        const int row = r + (hi ? 8 : 0);
        const int base = (wave * ROWS_PER_WAVE + row) * (TILE * KNN) + sub * KNN;
#pragma unroll
        for (int k = 0; k < KNN; ++k) {
            mD[base + k] = bd[r][k];
            mI[base + k] = bi[r][k];
        }
    }
    __syncthreads();

    // ---- final merge: lanes 0-15 each own one row ----
    if (lane < 16) {
        const int row = lane;
        const int base = (wave * ROWS_PER_WAVE + row) * (TILE * KNN);
        float fd[KNN];
        int   fi[KNN];
#pragma unroll
        for (int k = 0; k < KNN; ++k) { fd[k] = 3.0e38f; fi[k] = 0; }

        for (int j = 0; j < TILE * KNN; ++j) {
            const float v = mD[base + j];
            const int   c = mI[base + j];
            const bool c0 = v < fd[0];
            const bool c1 = v < fd[1];
            const bool c2 = v < fd[2];
            const bool c3 = v < fd[3];
            fd[3] = c2 ? fd[2] : (c3 ? v : fd[3]);
            fi[3] = c2 ? fi[2] : (c3 ? c : fi[3]);
            fd[2] = c1 ? fd[1] : (c2 ? v : fd[2]);
            fi[2] = c1 ? fi[1] : (c2 ? c : fi[2]);
            fd[1] = c0 ? fd[0] : (c1 ? v : fd[1]);
            fi[1] = c0 ? fi[0] : (c1 ? c : fi[1]);
            fd[0] = c0 ? v : fd[0];
            fi[0] = c0 ? c : fi[0];
        }

        // neighbor mean over the 4 selected points (full xyz, from L2)
        const int q = qbase + row;
        float gx = 0.f, gy = 0.f, gz = 0.f;
#pragma unroll
        for (int k = 0; k < KNN; ++k) {
            const float* p = xb + (size_t)fi[k] * 3;
            gx += p[0]; gy += p[1]; gz += p[2];
        }
        gx *= 0.25f; gy *= 0.25f; gz *= 0.25f;

        const float f0 = xb[q * 3 + 0] - gx;
        const float f1 = xb[q * 3 + 1] - gy;
        const float f2 = xb[q * 3 + 2] - gz;

        float* tout = trans + ((size_t)b * N_PTS + q) * 3;
#pragma unroll
        for (int cc = 0; cc < 3; ++cc) {
            tout[cc] = bias[cc] + Wm[cc * 3 + 0] * f0
                                + Wm[cc * 3 + 1] * f1
                                + Wm[cc * 3 + 2] * f2;
        }
    }
}

// ---------------------------------------------------------------------------
// Kernel 2: per-point normalization over axes (batch, channel): 24 values/pt,
// then gamma/beta, leaky-relu, residual add.
// ---------------------------------------------------------------------------
__global__ __launch_bounds__(256) void norm_kernel(
    const float* __restrict__ xyz,
    const float* __restrict__ trans,
    const float* __restrict__ gamma,
    const float* __restrict__ beta,
    float* __restrict__ out)
{
    const int n = blockIdx.x * blockDim.x + threadIdx.x;
    if (n >= N_PTS) return;

    float s = 0.f, ss = 0.f;
#pragma unroll
    for (int b = 0; b < N_BATCH; ++b) {
        const float* t = trans + ((size_t)b * N_PTS + n) * 3;
        const float t0 = t[0], t1 = t[1], t2 = t[2];
        s  += t0 + t1 + t2;
        ss += t0 * t0 + t1 * t1 + t2 * t2;
    }
    const float mean = s * (1.0f / 24.0f);
    const float var  = ss * (1.0f / 24.0f) - mean * mean;
    const float inv  = rsqrtf(var + EPSV);
    const float g  = gamma[n];
    const float be = beta[n];

#pragma unroll
    for (int b = 0; b < N_BATCH; ++b) {
        const size_t base = ((size_t)b * N_PTS + n) * 3;
#pragma unroll
        for (int cc = 0; cc < 3; ++cc) {
            const float t  = trans[base + cc];
            const float nm = (t - mean) * inv * g + be;
            const float act = nm > 0.f ? nm : SLOPE * nm;
            out[base + cc] = xyz[base + cc] + act;
        }
    }
}

// ---------------------------------------------------------------------------
extern "C" void kernel_launch(void* const* d_in, const int* in_sizes, int n_in,
                              void* d_out, int out_size, void* d_ws, size_t ws_size,
                              hipStream_t stream) {
    const float* xyz   = (const float*)d_in[0];
    const float* Wm    = (const float*)d_in[1];
    const float* bias  = (const float*)d_in[2];
    const float* gamma = (const float*)d_in[3];
    const float* beta  = (const float*)d_in[4];
    float* out   = (float*)d_out;
    float* trans = (float*)d_ws;   // B*N*3 floats = 393,216 bytes

    dim3 grid1(N_PTS / ROWS_PER_BLOCK, N_BATCH);  // (64, 8)
    knn_feat_kernel<<<grid1, 128, 0, stream>>>(xyz, Wm, bias, trans);

    norm_kernel<<<(N_PTS + 255) / 256, 256, 0, stream>>>(xyz, trans, gamma, beta, out);
}